// ConceptNet_New_70385924047534
// MI455X (gfx1250) — compile-verified
//
#include <hip/hip_runtime.h>
#include <hip/hip_bf16.h>
#include <math.h>

// Problem dimensions (fixed by the reference)
#define BSZ    4096
#define DD     768
#define CC     128
#define NCLUSTN 500
#define CSIZE  100
#define NTRAIN 100000
#define NCLS   10
#define KNN    10
#define EPSF   1e-12f
#define NBLK   782          // ceil(NTRAIN/128)
#define OUT_SC 40960        // BSZ*NCLS ; scalars live at OUT_SC..OUT_SC+3

// KNN tile staging
#define KCH    32           // K-chunk staged per TDM transfer
#define NCH    (DD / KCH)   // 24 chunks
#define TROW   34           // padded LDS row stride in floats (32 data + 2 pad)
#define TBUF   (128 * TROW) // floats per staging buffer (17408 B)

typedef __attribute__((ext_vector_type(2))) float v2f;
typedef __attribute__((ext_vector_type(8))) float v8f;
typedef __attribute__((ext_vector_type(4))) unsigned int u32x4;
typedef __attribute__((ext_vector_type(4))) int i32x4;
typedef __attribute__((ext_vector_type(8))) int i32x8;

#if defined(__has_builtin)
#if __has_builtin(__builtin_amdgcn_tensor_load_to_lds) && __has_builtin(__builtin_amdgcn_s_wait_tensorcnt)
#define HAVE_TDM 1
#endif
#endif

// D = A(16x4,f32) * B(4x16,f32) + C(16x16,f32), wave32.
__device__ __forceinline__ v8f wmma4(v2f a, v2f b, v8f c) {
#if __has_builtin(__builtin_amdgcn_wmma_f32_16x16x4_f32)
  return __builtin_amdgcn_wmma_f32_16x16x4_f32(false, a, false, b, (short)0, c, false, false);
#else
  c[0] += a[0]*b[0] + a[1]*b[1];
  return c;
#endif
}

__device__ __forceinline__ void zero_acc(v8f (&acc)[8]) {
#pragma unroll
  for (int i = 0; i < 8; ++i)
#pragma unroll
    for (int j = 0; j < 8; ++j) acc[i][j] = 0.0f;
}

#ifdef HAVE_TDM
// TDM: DMA a (tile_rows x 32) f32 tile (row stride 768 elems) from global into
// LDS at lds_off, inserting 2 pad DWORDs per 32 data DWORDs (row stride 34 floats).
// D# layout per CDNA5 ISA 8.3/8.4 (2-D tensor, groups 2/3 unused).
__device__ __forceinline__ void tdm_load_tile(unsigned lds_off, const float* gbase,
                                              unsigned long long elem_off,
                                              unsigned tile_rows) {
  const unsigned long long ga = (unsigned long long)(size_t)gbase + elem_off * 4ull;
  u32x4 g0;
  g0[0] = 1u;                                   // count=1, user descriptor
  g0[1] = lds_off;                              // lds_addr (bytes)
  g0[2] = (unsigned)(ga & 0xffffffffu);         // global_addr[31:0]
  g0[3] = (unsigned)((ga >> 32) & 0x01ffffffu)  // global_addr[56:32]
        | 0x80000000u;                          // type=2 ("image")
  i32x8 g1;
  g1[0] = (int)((2u << 16)                      // data_size = 4B
        | (1u << 20)                            // pad_enable
        | (4u << 22)                            // pad_interval: 32 DWORDs
        | (1u << 25));                          // pad_amount: 2 DWORDs
  g1[1] = (int)((DD & 0xffffu) << 16);          // tensor_dim0[15:0] (=768)
  g1[2] = (int)(((unsigned)DD >> 16) | ((NTRAIN & 0xffffu) << 16)); // dim0 hi | dim1 lo
  g1[3] = (int)(((unsigned)NTRAIN >> 16) | (((unsigned)KCH) << 16)); // dim1 hi | tile_dim0=32
  g1[4] = (int)(tile_rows & 0xffffu);           // tile_dim1 ; tile_dim2=0
  g1[5] = DD;                                   // tensor_dim0_stride lo32 (=768)
  g1[6] = 0;                                    // stride hi | dim1_stride lo
  g1[7] = 0;
  const i32x4 z4 = {0, 0, 0, 0};
#if __clang_major__ >= 23
  const i32x8 z8 = {0, 0, 0, 0, 0, 0, 0, 0};
  __builtin_amdgcn_tensor_load_to_lds(g0, g1, z4, z4, z8, 0);
#else
  __builtin_amdgcn_tensor_load_to_lds(g0, g1, z4, z4, 0);
#endif
}
#endif

// Fused KNN: TDM-stage TE tile -> LDS, e_sq on the fly, WMMA dot tile,
// d2 = csq + esq - 2*dot, per-concept top-10 candidates.
// (Defined first so the disasm snippet shows the TDM + LDS hot loop.)
__global__ __launch_bounds__(256) void k_knn(const float* __restrict__ cpt,
                                             const float* __restrict__ TE,
                                             const float* __restrict__ G,
                                             float* __restrict__ cd2,
                                             float* __restrict__ cdot) {
  __shared__ __align__(16) float smem[2 * TBUF];  // staging; reused as sdot later
  __shared__ float sesq[128];
  const int tid = threadIdx.x;
  const int wave = tid >> 5, lane = tid & 31;
  const int l15 = lane & 15, kh = lane >> 4;
  const int n0 = blockIdx.x * 128;
  const int nrows = (NTRAIN - n0 < 128) ? (NTRAIN - n0) : 128;
  const int mbase = wave * 16;
  v8f acc[8]; zero_acc(acc);
  float esq_p = 0.f;
  const int erow = tid >> 1, ehalf = (tid & 1) * 16;

#ifdef HAVE_TDM
  if (tid < 32)
    tdm_load_tile((unsigned)(size_t)(void*)smem, TE,
                  (unsigned long long)n0 * DD, (unsigned)nrows);
#endif
  for (int c = 0; c < NCH; ++c) {
    const int kbase = c * KCH;
    float* tb = smem + (c & 1) * TBUF;
#ifdef HAVE_TDM
    if (tid < 32) __builtin_amdgcn_s_wait_tensorcnt(0);  // chunk c arrived
    __syncthreads();                                     // visible; prev buffer free
    if ((c + 1 < NCH) && tid < 32)
      tdm_load_tile((unsigned)(size_t)(void*)(smem + ((c + 1) & 1) * TBUF), TE,
                    (unsigned long long)n0 * DD + (unsigned long long)(kbase + KCH),
                    (unsigned)nrows);
#else
    __syncthreads();
    for (int i = tid; i < 128 * KCH; i += 256) {
      const int nl = i >> 5, kl = i & (KCH - 1);
      const int gn = n0 + nl;
      tb[nl * TROW + kl] = (gn < NTRAIN) ? TE[(size_t)gn * DD + kbase + kl] : 0.f;
    }
    __syncthreads();
#endif
    // fused e_sq partial (2 threads per staged row)
    {
      const float* rp = tb + erow * TROW + ehalf;
#pragma unroll
      for (int i = 0; i < 16; ++i) esq_p += rp[i] * rp[i];
    }
    // WMMA over this K chunk. No OOB guard on B: garbage rows only affect
    // output columns that the selection phase never reads (j < nrows).
    for (int kk = 0; kk < KCH; kk += 4) {
      const int kl = kk + 2 * kh;
      const int ka = kbase + kl;
      v2f a;
      a.x = cpt[ka * CC + mbase + l15];
      a.y = cpt[(ka + 1) * CC + mbase + l15];
#pragma unroll
      for (int ns = 0; ns < 8; ++ns) {
        const int nl = ns * 16 + l15;
        const v2f b = *(const v2f*)(tb + nl * TROW + kl);   // ds_load_b64
        acc[ns] = wmma4(a, b, acc[ns]);
      }
    }
  }
  // combine e_sq halves (lanes 2r / 2r+1 of same wave)
  esq_p += __shfl_down(esq_p, 1, 32);
  if ((tid & 1) == 0) sesq[erow] = esq_p;
  __syncthreads();

  // two-phase accumulator spill + per-row top-10 (sdot aliases staging buffers)
  float* sdot = smem;
  for (int half = 0; half < 2; ++half) {
    if ((wave >> 2) == half) {
      const int lrb = mbase - half * 64;
#pragma unroll
      for (int ns = 0; ns < 8; ++ns)
#pragma unroll
        for (int r = 0; r < 8; ++r)
          sdot[(lrb + r + kh * 8) * 130 + ns * 16 + l15] = acc[ns][r];
    }
    __syncthreads();
    if (tid < 64) {
      const int m = half * 64 + tid;
      const float csq = G[m * CC + m];
      float bd[KNN], bv[KNN];
#pragma unroll
      for (int i = 0; i < KNN; ++i) { bd[i] = __builtin_inff(); bv[i] = 0.f; }
      for (int j = 0; j < nrows; ++j) {
        const float dot = sdot[tid * 130 + j];
        const float d2v = csq + sesq[j] - 2.f * dot;
        if (d2v < bd[KNN - 1]) {
          int p = KNN - 1;
          while (p > 0 && bd[p - 1] > d2v) { bd[p] = bd[p - 1]; bv[p] = bv[p - 1]; --p; }
          bd[p] = d2v; bv[p] = dot;
        }
      }
      const size_t base = ((size_t)blockIdx.x * CC + m) * KNN;
#pragma unroll
      for (int i = 0; i < KNN; ++i) { cd2[base + i] = bd[i]; cdot[base + i] = bv[i]; }
    }
    __syncthreads();
  }
}

// ---------------- gram: G = concept.T @ concept (128x128, K=768) -------------
__global__ __launch_bounds__(256) void k_gram(const float* __restrict__ cpt,
                                              float* __restrict__ G) {
  const int tid = threadIdx.x;
  const int wave = tid >> 5, lane = tid & 31;
  const int l15 = lane & 15, kh = lane >> 4;
  const int mbase = wave * 16;
  v8f acc[8]; zero_acc(acc);
  for (int k0 = 0; k0 < DD; k0 += 4) {
    const int ka = k0 + 2 * kh;
    v2f a;
    a.x = cpt[ka * CC + mbase + l15];
    a.y = cpt[(ka + 1) * CC + mbase + l15];
#pragma unroll
    for (int ns = 0; ns < 8; ++ns) {
      const int n = ns * 16 + l15;
      v2f b;
      b.x = cpt[ka * CC + n];
      b.y = cpt[(ka + 1) * CC + n];
      acc[ns] = wmma4(a, b, acc[ns]);
    }
  }
#pragma unroll
  for (int ns = 0; ns < 8; ++ns)
#pragma unroll
    for (int r = 0; r < 8; ++r)
      G[(mbase + r + kh * 8) * CC + ns * 16 + l15] = acc[ns][r];
}

__global__ void k_cstats(const float* __restrict__ G, float* __restrict__ inv) {
  const int c = threadIdx.x;
  if (c < CC) inv[c] = 1.0f / fmaxf(sqrtf(G[c * CC + c]), EPSF);
}

__global__ void k_gram_reduce(const float* __restrict__ G, float* __restrict__ out) {
  __shared__ float rs[256], rt[256];
  const int tid = threadIdx.x;
  float s = 0.f, t = 0.f;
  for (int i = tid; i < CC * CC; i += 256) {
    const float v = G[i];
    s += v;
    if ((i % (CC + 1)) == 0) t += v;
  }
  rs[tid] = s; rt[tid] = t; __syncthreads();
  for (int off = 128; off > 0; off >>= 1) {
    if (tid < off) { rs[tid] += rs[tid + off]; rt[tid] += rt[tid + off]; }
    __syncthreads();
  }
  if (tid == 0) out[OUT_SC + 3] = (rs[0] - rt[0]) / (float)(CC * CC);
}

__global__ void k_cw(const float* __restrict__ cpt, const float* __restrict__ w,
                     float* __restrict__ Cw) {
  const int idx = blockIdx.x * 256 + threadIdx.x;
  if (idx >= CC * NCLS) return;
  const int c = idx / NCLS, j = idx % NCLS;
  float s = 0.f;
  for (int d = 0; d < DD; ++d) s += cpt[d * CC + c] * w[(size_t)j * DD + d];
  Cw[idx] = s;
}

// Gauss-Jordan solve of G Y = Cw (G SPD) in workspace memory.
__global__ __launch_bounds__(256) void k_solve(const float* __restrict__ G,
                                               const float* __restrict__ Cw,
                                               float* __restrict__ GS,
                                               float* __restrict__ Y) {
  const int tid = threadIdx.x;
  for (int i = tid; i < CC * CC; i += 256) GS[i] = G[i];
  for (int i = tid; i < CC * NCLS; i += 256) Y[i] = Cw[i];
  __syncthreads();
  for (int p = 0; p < CC; ++p) {
    if (tid < CC && tid != p) {
      const float f = GS[tid * CC + p] / GS[p * CC + p];
      for (int c2 = p; c2 < CC; ++c2) GS[tid * CC + c2] -= f * GS[p * CC + c2];
      for (int j = 0; j < NCLS; ++j) Y[tid * NCLS + j] -= f * Y[p * NCLS + j];
    }
    __syncthreads();
  }
  if (tid < CC) {
    const float ip = 1.0f / GS[tid * CC + tid];
    for (int j = 0; j < NCLS; ++j) Y[tid * NCLS + j] *= ip;
  }
}

// Z = train_embedding @ concept   (4096x128, K=768)
__global__ __launch_bounds__(256) void k_gemm_z(const float* __restrict__ temb,
                                                const float* __restrict__ cpt,
                                                float* __restrict__ Z) {
  const int tid = threadIdx.x;
  const int wave = tid >> 5, lane = tid & 31;
  const int l15 = lane & 15, kh = lane >> 4;
  const int m0 = blockIdx.x * 128 + wave * 16;
  v8f acc[8]; zero_acc(acc);
  const float* ap = temb + (size_t)(m0 + l15) * DD;
  for (int k0 = 0; k0 < DD; k0 += 4) {
    const int ka = k0 + 2 * kh;
    const v2f a = *(const v2f*)(ap + ka);
#pragma unroll
    for (int ns = 0; ns < 8; ++ns) {
      const int n = ns * 16 + l15;
      v2f b;
      b.x = cpt[ka * CC + n];
      b.y = cpt[(ka + 1) * CC + n];
      acc[ns] = wmma4(a, b, acc[ns]);
    }
  }
#pragma unroll
  for (int ns = 0; ns < 8; ++ns)
#pragma unroll
    for (int r = 0; r < 8; ++r)
      Z[(size_t)(m0 + r + kh * 8) * CC + ns * 16 + l15] = acc[ns][r];
}

__global__ void k_ypred(const float* __restrict__ Z, const float* __restrict__ Y,
                        const float* __restrict__ b, float* __restrict__ out) {
  const int idx = blockIdx.x * 256 + threadIdx.x;
  if (idx >= BSZ * NCLS) return;
  const int i = idx / NCLS, j = idx % NCLS;
  float s = b[j];
  const float* z = Z + (size_t)i * CC;
  for (int c = 0; c < CC; ++c) s += z[c] * Y[c * NCLS + j];
  out[idx] = s;
}

__global__ void k_cmean(const float* __restrict__ clus, float* __restrict__ CM) {
  const int idx = blockIdx.x * 256 + threadIdx.x;
  if (idx >= NCLUSTN * DD) return;
  const int n = idx / DD, d = idx % DD;
  const float* p = clus + (size_t)n * (CSIZE * DD) + d;
  float s = 0.f;
  for (int ss = 0; ss < CSIZE; ++ss) s += p[ss * DD];
  CM[idx] = s * (1.0f / CSIZE);
}

// S = |cluster_mean @ concept| * inv_colnorm(concept)
__global__ __launch_bounds__(256) void k_gemm_s(const float* __restrict__ CM,
                                                const float* __restrict__ cpt,
                                                const float* __restrict__ inv,
                                                float* __restrict__ S) {
  const int tid = threadIdx.x;
  const int wave = tid >> 5, lane = tid & 31;
  const int l15 = lane & 15, kh = lane >> 4;
  const int m0 = blockIdx.x * 128 + wave * 16;
  v8f acc[8]; zero_acc(acc);
  const int gm = m0 + l15;
  const bool okm = gm < NCLUSTN;
  const float* ap = CM + (size_t)(okm ? gm : 0) * DD;
  for (int k0 = 0; k0 < DD; k0 += 4) {
    const int ka = k0 + 2 * kh;
    v2f a = {0.f, 0.f};
    if (okm) a = *(const v2f*)(ap + ka);
#pragma unroll
    for (int ns = 0; ns < 8; ++ns) {
      const int n = ns * 16 + l15;
      v2f b;
      b.x = cpt[ka * CC + n];
      b.y = cpt[(ka + 1) * CC + n];
      acc[ns] = wmma4(a, b, acc[ns]);
    }
  }
#pragma unroll
  for (int ns = 0; ns < 8; ++ns)
#pragma unroll
    for (int r = 0; r < 8; ++r) {
      const int m = m0 + r + kh * 8;
      const int n = ns * 16 + l15;
      if (m < NCLUSTN) S[m * CC + n] = fabsf(acc[ns][r]) * inv[n];
    }
}

__global__ void k_colnorm(const float* __restrict__ S, float* __restrict__ cn) {
  const int c = threadIdx.x;
  if (c >= CC) return;
  float s = 0.f;
  for (int m = 0; m < NCLUSTN; ++m) { const float v = S[m * CC + c]; s += v * v; }
  cn[c] = 1.0f / fmaxf(sqrtf(s), EPSF);
}

__global__ void k_scorestats(const float* __restrict__ S, const float* __restrict__ cn,
                             float* __restrict__ out) {
  __shared__ float r1[256], r2[256], r3[256];
  const int tid = threadIdx.x;
  float s1 = 0.f, ssq = 0.f, srw = 0.f;
  for (int m = tid; m < NCLUSTN; m += 256) {
    float rs = 0.f;
    for (int c = 0; c < CC; ++c) {
      const float v = S[m * CC + c] * cn[c];
      s1 += v; ssq += v * v; rs += v;
    }
    srw += rs * rs;
  }
  r1[tid] = s1; r2[tid] = ssq; r3[tid] = srw; __syncthreads();
  for (int off = 128; off > 0; off >>= 1) {
    if (tid < off) { r1[tid] += r1[tid + off]; r2[tid] += r2[tid + off]; r3[tid] += r3[tid + off]; }
    __syncthreads();
  }
  if (tid == 0) { out[OUT_SC + 0] = r1[0]; out[OUT_SC + 1] = r3[0] - r2[0]; }
}

// Merge per-block candidates -> L_sparse_1_new
__global__ void k_merge(const float* __restrict__ cd2, const float* __restrict__ cdot,
                        float* __restrict__ out) {
  __shared__ float red[128];
  const int c = threadIdx.x;
  float bd[KNN], bv[KNN];
#pragma unroll
  for (int i = 0; i < KNN; ++i) { bd[i] = __builtin_inff(); bv[i] = 0.f; }
  for (int b = 0; b < NBLK; ++b) {
    const size_t base = ((size_t)b * CC + c) * KNN;
    for (int j = 0; j < KNN; ++j) {
      const float d2v = cd2[base + j];
      if (d2v >= bd[KNN - 1]) break;  // per-block candidates sorted ascending
      const float dv = cdot[base + j];
      int p = KNN - 1;
      while (p > 0 && bd[p - 1] > d2v) { bd[p] = bd[p - 1]; bv[p] = bv[p - 1]; --p; }
      bd[p] = d2v; bv[p] = dv;
    }
  }
  float s = 0.f;
#pragma unroll
  for (int i = 0; i < KNN; ++i) s += bv[i];
  red[c] = s; __syncthreads();
  for (int off = 64; off > 0; off >>= 1) {
    if (c < off) red[c] += red[c + off];
    __syncthreads();
  }
  if (c == 0) out[OUT_SC + 2] = red[0] / (float)(KNN * CC);
}

extern "C" void kernel_launch(void* const* d_in, const int* in_sizes, int n_in,
                              void* d_out, int out_size, void* d_ws, size_t ws_size,
                              hipStream_t stream) {
  (void)in_sizes; (void)n_in; (void)out_size; (void)ws_size;
  const float* temb = (const float*)d_in[0];   // (4096,768)
  const float* cpt  = (const float*)d_in[1];   // (768,128)
  const float* clus = (const float*)d_in[2];   // (500,100,768)
  const float* TE   = (const float*)d_in[3];   // (100000,768)
  const float* w    = (const float*)d_in[4];   // (10,768)
  const float* bh   = (const float*)d_in[5];   // (10,)
  float* out = (float*)d_out;
  float* ws  = (float*)d_ws;

  float* G    = ws + 0;        // 16384
  float* Cw   = ws + 16384;    // 1280
  float* Y    = ws + 17664;    // 1280
  float* inv  = ws + 18944;    // 128
  float* Z    = ws + 19072;    // 524288
  float* CM   = ws + 543360;   // 384000
  float* S    = ws + 927360;   // 64000
  float* cn   = ws + 991360;   // 128
  float* cd2  = ws + 1091488;  // 1000960
  float* cdot = ws + 2092448;  // 1000960
  float* GS   = ws + 3093408;  // 16384  (total ~12.4 MB)

  k_gram<<<1, 256, 0, stream>>>(cpt, G);
  k_cstats<<<1, 128, 0, stream>>>(G, inv);
  k_gram_reduce<<<1, 256, 0, stream>>>(G, out);
  k_cw<<<5, 256, 0, stream>>>(cpt, w, Cw);
  k_solve<<<1, 256, 0, stream>>>(G, Cw, GS, Y);
  k_gemm_z<<<32, 256, 0, stream>>>(temb, cpt, Z);
  k_ypred<<<160, 256, 0, stream>>>(Z, Y, bh, out);
  k_cmean<<<1500, 256, 0, stream>>>(clus, CM);
  k_gemm_s<<<4, 256, 0, stream>>>(CM, cpt, inv, S);
  k_colnorm<<<1, 128, 0, stream>>>(S, cn);
  k_scorestats<<<1, 256, 0, stream>>>(S, cn, out);
  k_knn<<<NBLK, 256, 0, stream>>>(cpt, TE, G, cd2, cdot);
  k_merge<<<1, 128, 0, stream>>>(cd2, cdot, out);
}